// GraphCAD_73521250173055
// MI455X (gfx1250) — compile-verified
//
#include <hip/hip_runtime.h>
#include <hip/hip_bf16.h>

// ---------------------------------------------------------------------------
// GraphCAD pipeline for MI455X (gfx1250, wave32).
//
// Structural optimization: C2 == 1  =>  s1 = softmax over 1 col = ones.
//   x_cov2[f]   = sum_c x_cov1[c,f]
//   g0[c,f]     = 2*sigmoid(feature_corr[f] - x_cov2[f])  = gvec[f]  (rank-0 in c)
//   m[e,f]      = dot(s0[r_e], s0[c_e]) * gvec[f]
//   adjs[e,f]   = adj_vals[e] * d_e^(f+1) * cumprod(gvec)[f]
// The [E,100]@[100,64] edge GEMM collapses; the edge stage is a streaming
// 160MB write (HBM roofline ~7us at 23.3 TB/s) + L2-resident s0 gathers.
//
// Dense GEMMs use V_WMMA_F32_16X16X4_F32, templated so the k-loop fully
// unrolls with no guards: B staged in LDS in per-lane WMMA layout, each wave
// computes a 16x32 tile (2 accumulators reuse each A fetch).
// ---------------------------------------------------------------------------

#define NN   10000
#define FF   64
#define HH   128
#define EE   640000
#define CC1  100

typedef float v2f __attribute__((ext_vector_type(2)));
typedef float v4f __attribute__((ext_vector_type(4)));
typedef float v8f __attribute__((ext_vector_type(8)));

// ------------------------- BatchNorm -------------------------

__global__ void bn_stats_kernel(const float* __restrict__ x,
                                const float* __restrict__ gamma,
                                const float* __restrict__ beta,
                                float* __restrict__ bn_a,
                                float* __restrict__ bn_b) {
  __shared__ float s1[256], s2[256];
  const int f = blockIdx.x;      // feature column, 64 blocks
  const int t = threadIdx.x;     // 256 threads
  float s = 0.f, q = 0.f;
  for (int i = t; i < NN; i += 256) {
    float v = x[(size_t)i * FF + f];
    s += v; q += v * v;
  }
  s1[t] = s; s2[t] = q; __syncthreads();
  for (int st = 128; st > 0; st >>= 1) {
    if (t < st) { s1[t] += s1[t + st]; s2[t] += s2[t + st]; }
    __syncthreads();
  }
  if (t == 0) {
    float mu  = s1[0] / (float)NN;
    float var = s2[0] / (float)NN - mu * mu;     // population var (ddof=0)
    float rstd = rsqrtf(var + 1e-5f);
    float a = gamma[f] * rstd;
    bn_a[f] = a;
    bn_b[f] = beta[f] - mu * a;
  }
}

__global__ void bn_apply_kernel(const float* __restrict__ x,
                                const float* __restrict__ bn_a,
                                const float* __restrict__ bn_b,
                                float* __restrict__ x0) {
  int idx = blockIdx.x * blockDim.x + threadIdx.x;   // N*F = 640000 exact
  int f = idx & (FF - 1);
  x0[idx] = x[idx] * bn_a[f] + bn_b[f];
}

// ------------------------- WMMA fp32 GEMM -------------------------
// C[10000,NCOLS] = act(A[10000,KD] @ B[KD,NCOLS] + bias)
// Block: 160 threads = 5 waves, covers 80 rows x 32 cols.
//   grid = (125, ceil(NCOLS/32))
// B tile is staged in LDS in the exact per-lane WMMA operand layout
// [step][lane][4] (zero-padded past NCOLS), so the unrolled inner loop is
// 1x global_load_b64 (A) + 1x ds_load_b128 (B) + 2x v_wmma per k-step with
// no divergence (EXEC all-ones around every WMMA, per ISA requirement).
// ACT: 0=none, 1=relu, 2=prelu(a)

template<int KD, int NCOLS, int ACT>
__global__ __launch_bounds__(160)
void gemm_wmma_kernel(const float* __restrict__ A,
                      const float* __restrict__ B,
                      const float* __restrict__ bias,
                      float* __restrict__ C,
                      const float* __restrict__ prelu_a) {
  constexpr int STEPS = KD / 4;
  __shared__ v4f Bsh4[STEPS * 32];            // [step][lane] -> {b0.x,b0.y,b1.x,b1.y}
  float* Bsh = (float*)Bsh4;
  const int t = threadIdx.x;
  const int colbase = blockIdx.y * 32;

  // ---- cooperative stage of B into WMMA per-lane layout ----
  for (int p = t; p < STEPS * 128; p += 160) {
    const int s  = p >> 7;           // k-step
    const int l  = (p >> 2) & 31;    // consuming lane
    const int j  = p & 3;            // 0,1 -> colA pair; 2,3 -> colB pair
    const int hf = l >> 4;
    const int id = l & 15;
    const int kk = s * 4 + 2 * hf + (j & 1);
    const int col = colbase + ((j >> 1) * 16) + id;
    float v = 0.0f;
    if ((NCOLS % 32 == 0) || (col < NCOLS)) v = B[(size_t)kk * NCOLS + col];
    Bsh[p] = v;
  }
  __syncthreads();

  const int wave = t >> 5;
  const int lane = t & 31;
  const int half = lane >> 4;                 // ISA A/B layout: half 0 -> k+{0,1}, half 1 -> k+{2,3}
  const int idx  = lane & 15;
  const int row0 = blockIdx.x * 80 + wave * 16;
  const int colA = colbase + idx;
  const int colB = colA + 16;

  v8f acc0 = {}, acc1 = {};
  const float* Arow = A + (size_t)(row0 + idx) * KD + 2 * half;
  #pragma unroll
  for (int s = 0; s < STEPS; ++s) {
    v2f a;
    a.x = Arow[s * 4 + 0];
    a.y = Arow[s * 4 + 1];
    v4f bv = Bsh4[s * 32 + lane];
    v2f b0; b0.x = bv.x; b0.y = bv.y;
    v2f b1; b1.x = bv.z; b1.y = bv.w;
    acc0 = __builtin_amdgcn_wmma_f32_16x16x4_f32(false, a, false, b0, (short)0, acc0, false, false);
    acc1 = __builtin_amdgcn_wmma_f32_16x16x4_f32(false, a, false, b1, (short)0, acc1, false, false);
  }

  const bool okA = (NCOLS % 32 == 0) || (colA < NCOLS);
  const bool okB = (NCOLS % 32 == 0) || (colB < NCOLS);
  const float pa = (ACT == 2) ? prelu_a[0] : 0.0f;
  const float bA = okA ? bias[colA] : 0.0f;
  const float bB = okB ? bias[colB] : 0.0f;
  #pragma unroll
  for (int j = 0; j < 8; ++j) {
    const int r = row0 + j + 8 * half;        // ISA C/D layout
    float v0 = acc0[j] + bA;
    float v1 = acc1[j] + bB;
    if (ACT == 1) { v0 = fmaxf(v0, 0.f); v1 = fmaxf(v1, 0.f); }
    if (ACT == 2) { v0 = v0 > 0.f ? v0 : pa * v0; v1 = v1 > 0.f ? v1 : pa * v1; }
    if (okA) C[(size_t)r * NCOLS + colA] = v0;
    if (okB) C[(size_t)r * NCOLS + colB] = v1;
  }
}

// ------------------------- softmax over 100 cols (in place) -------------------------

__global__ void softmax_rows_kernel(float* __restrict__ s) {
  __shared__ float red[128];
  const int i = blockIdx.x;
  const int t = threadIdx.x;
  float* row = s + (size_t)i * CC1;
  float m = -1e30f;
  for (int c = t; c < CC1; c += 128) m = fmaxf(m, row[c]);
  red[t] = m; __syncthreads();
  for (int st = 64; st > 0; st >>= 1) {
    if (t < st) red[t] = fmaxf(red[t], red[t + st]);
    __syncthreads();
  }
  m = red[0]; __syncthreads();
  float sum = 0.f;
  for (int c = t; c < CC1; c += 128) { float e = __expf(row[c] - m); row[c] = e; sum += e; }
  red[t] = sum; __syncthreads();
  for (int st = 64; st > 0; st >>= 1) {
    if (t < st) red[t] += red[t + st];
    __syncthreads();
  }
  const float inv = 1.0f / red[0];
  for (int c = t; c < CC1; c += 128) row[c] *= inv;
}

// ------------------------- x_cov1 = s0^T @ x_cov  (split-K, LDS accumulate) ----------

__global__ void zero_kernel(float* __restrict__ p, int n) {
  int i = blockIdx.x * blockDim.x + threadIdx.x;
  if (i < n) p[i] = 0.0f;
}

__global__ void xcov1_accum_kernel(const float* __restrict__ s0,
                                   const float* __restrict__ xcov,
                                   float* __restrict__ xcov1,
                                   int rows_per_block) {
  __shared__ float acc[CC1 * FF];   // 25.6 KB
  __shared__ float srow[CC1];
  __shared__ float xrow[FF];
  const int t = threadIdx.x;        // 256
  for (int p = t; p < CC1 * FF; p += 256) acc[p] = 0.0f;
  __syncthreads();
  const int r0 = blockIdx.x * rows_per_block;
  int r1 = r0 + rows_per_block; if (r1 > NN) r1 = NN;
  for (int i = r0; i < r1; ++i) {
    if (t < CC1) srow[t] = s0[(size_t)i * CC1 + t];
    else if (t >= 128 && t < 128 + FF) xrow[t - 128] = xcov[(size_t)i * FF + (t - 128)];
    __syncthreads();
    for (int p = t; p < CC1 * FF; p += 256) {
      int c = p >> 6, f = p & (FF - 1);
      acc[p] += srow[c] * xrow[f];
    }
    __syncthreads();
  }
  for (int p = t; p < CC1 * FF; p += 256) atomicAdd(&xcov1[p], acc[p]);
}

// ------------------------- gvec + cumprod(gvec)  (s1 == ones, C2 == 1) --------------

__global__ void gcum_kernel(const float* __restrict__ xcov1,
                            const float* __restrict__ fc,
                            float* __restrict__ cumg) {
  __shared__ float g[FF];
  const int f = threadIdx.x;   // 64 threads
  float s = 0.f;
  for (int c = 0; c < CC1; ++c) s += xcov1[c * FF + f];
  float z = fc[f] - s;                          // (s1@fc - corr1)[.,f]
  g[f] = 2.0f / (1.0f + __expf(-z));            // 2*sigmoid
  __syncthreads();
  if (f == 0) { float cp = 1.0f; for (int j = 0; j < FF; ++j) { cp *= g[j]; g[j] = cp; } }
  __syncthreads();
  cumg[f] = g[f];
}

// ------------------------- adjs: wave per edge, streaming 160MB write ---------------

__global__ void edge_adjs_kernel(const int* __restrict__ rows,
                                 const int* __restrict__ cols,
                                 const float* __restrict__ adjv,
                                 const float* __restrict__ s0,
                                 const float* __restrict__ cumg,
                                 float* __restrict__ out) {
  __shared__ float cg[FF];
  const int t = threadIdx.x;
  if (t < FF) cg[t] = cumg[t];
  __syncthreads();
  const int lane = t & 31;
  const int e = (blockIdx.x * blockDim.x + t) >> 5;
  if (e >= EE) return;
  const int r = rows[e], c = cols[e];
  const float* sr = s0 + (size_t)r * CC1;
  const float* sc = s0 + (size_t)c * CC1;
  float d = 0.f;
  for (int cc = lane; cc < CC1; cc += 32) d += sr[cc] * sc[cc];
  for (int off = 16; off > 0; off >>= 1) d += __shfl_xor(d, off, 32);
  const float av = adjv[e];
  const float ld = __logf(d);                    // d > 0 (softmax dot)
  float* o = out + (size_t)e * FF;
  o[lane]      = av * cg[lane]      * __expf((float)(lane + 1)  * ld);
  o[lane + 32] = av * cg[lane + 32] * __expf((float)(lane + 33) * ld);
}

// ------------------------- SpMM: out[r,:] += v * x[c,:]  (wave per edge) ------------

__global__ void spmm_kernel(const int* __restrict__ rows,
                            const int* __restrict__ cols,
                            const float* __restrict__ vals,
                            const float* __restrict__ xin,
                            float* __restrict__ xout) {
  const int lane = threadIdx.x & 31;
  const int e = (blockIdx.x * blockDim.x + threadIdx.x) >> 5;
  if (e >= EE) return;
  const int r = rows[e], c = cols[e];
  const float v = vals[e];
  atomicAdd(&xout[(size_t)r * FF + lane],      v * xin[(size_t)c * FF + lane]);
  atomicAdd(&xout[(size_t)r * FF + lane + 32], v * xin[(size_t)c * FF + lane + 32]);
}

// ------------------------- final 128->2 head + log_softmax -------------------------

__global__ void mlp3_logsoftmax_kernel(const float* __restrict__ h,
                                       const float* __restrict__ W3,
                                       const float* __restrict__ b3,
                                       float* __restrict__ out) {
  const int i = blockIdx.x * blockDim.x + threadIdx.x;
  if (i >= NN) return;
  const float* hr = h + (size_t)i * HH;
  float z0 = b3[0], z1 = b3[1];
  #pragma unroll 8
  for (int k = 0; k < HH; ++k) { float v = hr[k]; z0 += v * W3[2 * k]; z1 += v * W3[2 * k + 1]; }
  const float m = fmaxf(z0, z1);
  const float lse = m + __logf(__expf(z0 - m) + __expf(z1 - m));
  out[2 * i]     = z0 - lse;
  out[2 * i + 1] = z1 - lse;
}

// ------------------------- launch -------------------------

extern "C" void kernel_launch(void* const* d_in, const int* in_sizes, int n_in,
                              void* d_out, int out_size, void* d_ws, size_t ws_size,
                              hipStream_t stream) {
  const float* x     = (const float*)d_in[0];
  const float* xcov  = (const float*)d_in[1];
  const int*   ei    = (const int*)d_in[2];
  const int*   rows  = ei;
  const int*   cols  = ei + EE;
  const float* adjv  = (const float*)d_in[3];
  const float* normv = (const float*)d_in[4];
  const float* fc    = (const float*)d_in[5];
  const float* gamma = (const float*)d_in[6];
  const float* beta  = (const float*)d_in[7];
  const float* p0W1  = (const float*)d_in[8];
  const float* p0b1  = (const float*)d_in[9];
  const float* p0W2  = (const float*)d_in[10];
  const float* p0b2  = (const float*)d_in[11];
  // pool1_* (d_in[12..15]) are mathematically dead: softmax over C2==1 => s1==1.
  const float* mW1   = (const float*)d_in[16];
  const float* mb1   = (const float*)d_in[17];
  const float* mW2   = (const float*)d_in[18];
  const float* mb2   = (const float*)d_in[19];
  const float* mW3   = (const float*)d_in[20];
  const float* mb3   = (const float*)d_in[21];
  const float* pa    = (const float*)d_in[22];

  float* W = (float*)d_ws;
  float* bn_a  = W;                 // 64
  float* bn_b  = W + 64;            // 64
  float* cumg  = W + 128;           // 64
  float* xcov1 = W + 256;           // 6400
  float* xA    = W + 16384;         // 640000
  float* xB    = xA + (size_t)NN * FF;       // 640000
  float* hbuf  = xB + (size_t)NN * FF;       // 1280000
  float* h2buf = hbuf + (size_t)NN * HH;     // 1280000
  float* s0    = h2buf + (size_t)NN * HH;    // 1000000  (total ~19.4 MB)

  float* out_ls   = (float*)d_out;           // [10000,2]
  float* out_adjs = out_ls + 2 * NN;         // [E,64]

  const dim3 gemm_blk(160);                  // 5 waves = 80 rows per block
  const dim3 gemm_grid(NN / 80, 4);          // 4 col-tiles of 32 (128 or padded 100)

  // 1) BatchNorm -> xA (= x0, propagation seed; alpha=0 so x0 unused afterwards)
  bn_stats_kernel<<<FF, 256, 0, stream>>>(x, gamma, beta, bn_a, bn_b);
  bn_apply_kernel<<<(NN * FF) / 256, 256, 0, stream>>>(x, bn_a, bn_b, xA);

  // 2) cluster-0: h = relu(xcov @ p0W1 + b1); s0 = softmax(h @ p0W2 + b2)
  gemm_wmma_kernel<FF, HH, 1><<<gemm_grid, gemm_blk, 0, stream>>>(xcov, p0W1, p0b1, hbuf, nullptr);
  gemm_wmma_kernel<HH, CC1, 0><<<gemm_grid, gemm_blk, 0, stream>>>(hbuf, p0W2, p0b2, s0, nullptr);
  softmax_rows_kernel<<<NN, 128, 0, stream>>>(s0);

  // 3) x_cov1 = s0^T @ xcov -> colsum -> gvec -> cumprod
  zero_kernel<<<(CC1 * FF + 255) / 256, 256, 0, stream>>>(xcov1, CC1 * FF);
  xcov1_accum_kernel<<<100, 256, 0, stream>>>(s0, xcov, xcov1, NN / 100);
  gcum_kernel<<<1, FF, 0, stream>>>(xcov1, fc, cumg);

  // 4) adjs[e,f] = adj[e] * d_e^(f+1) * cumg[f]   (dominant: 160MB streaming write)
  edge_adjs_kernel<<<(EE * 32) / 256, 256, 0, stream>>>(rows, cols, adjv, s0, cumg, out_adjs);

  // 5) propagation x = norm_adj @ x, K=3  (xA -> xB -> xA -> xB)
  zero_kernel<<<(NN * FF) / 256, 256, 0, stream>>>(xB, NN * FF);
  spmm_kernel<<<(EE * 32) / 256, 256, 0, stream>>>(rows, cols, normv, xA, xB);
  zero_kernel<<<(NN * FF) / 256, 256, 0, stream>>>(xA, NN * FF);
  spmm_kernel<<<(EE * 32) / 256, 256, 0, stream>>>(rows, cols, normv, xB, xA);
  zero_kernel<<<(NN * FF) / 256, 256, 0, stream>>>(xB, NN * FF);
  spmm_kernel<<<(EE * 32) / 256, 256, 0, stream>>>(rows, cols, normv, xA, xB);

  // 6) MLP head with PReLU, then log_softmax
  gemm_wmma_kernel<FF, HH, 2><<<gemm_grid, gemm_blk, 0, stream>>>(xB, mW1, mb1, hbuf, pa);
  gemm_wmma_kernel<HH, HH, 2><<<gemm_grid, gemm_blk, 0, stream>>>(hbuf, mW2, mb2, h2buf, pa);
  mlp3_logsoftmax_kernel<<<(NN + 255) / 256, 256, 0, stream>>>(h2buf, mW3, mb3, out_ls);

  (void)in_sizes; (void)n_in; (void)out_size; (void)ws_size;
}